// MFGCGRU_23570780521075
// MI455X (gfx1250) — compile-verified
//
#include <hip/hip_runtime.h>
#include <math.h>

typedef float v2f __attribute__((ext_vector_type(2)));
typedef float v8f __attribute__((ext_vector_type(8)));

#define N_NODES 1024
#define UNITS_  64
#define BATCH_  32
#define IN_DIM  2
#define FEAT    66          // IN_DIM + UNITS
#define NCOLS   2112        // BATCH_*FEAT
#define LDX     2176        // padded to 17*128
#define KROWS   264         // FEAT*4 supports
#define NROWS   32768       // BATCH_*N_NODES

// workspace layout (float offsets); total ~92 MB
#define OFF_Q      0u
#define OFF_K      65536u
#define OFF_ATTN   131072u
#define OFF_X      1179648u
#define OFF_Y0     3407872u
#define OFF_Y1     5636096u
#define OFF_Y2     7864320u
#define OFF_AST    10092544u
#define OFF_R      18743296u
#define OFF_U      20840448u
#define OFF_WP     22937600u   // 3 * 264 * 64 paired-layout weights

// ---------------- q/k projection: q = relu(feat @ Wq), k = relu(feat @ Wk) ----------------
__global__ void qk_kernel(const float* __restrict__ feat, const float* __restrict__ Wq,
                          const float* __restrict__ Wk, float* __restrict__ q, float* __restrict__ k) {
  int idx = blockIdx.x * blockDim.x + threadIdx.x;   // 1024*64
  int n = idx >> 6, u = idx & 63;
  const float* f = feat + n * 32;
  float aq = 0.f, ak = 0.f;
  #pragma unroll
  for (int j = 0; j < 32; ++j) {
    float fv = f[j];
    aq = fmaf(fv, Wq[j * 64 + u], aq);
    ak = fmaf(fv, Wk[j * 64 + u], ak);
  }
  q[idx] = fmaxf(aq, 0.f);
  k[idx] = fmaxf(ak, 0.f);
}

// ---------------- attention support: softmax_rows(q @ k^T / 8) ----------------
__global__ void attn_kernel(const float* __restrict__ q, const float* __restrict__ k,
                            float* __restrict__ attn) {
  __shared__ float qrow[64];
  __shared__ float sc[N_NODES];
  __shared__ float red[256];
  const int n = blockIdx.x, t = threadIdx.x;
  if (t < 64) qrow[t] = q[n * 64 + t];
  __syncthreads();
  float lmax = -3.402823e38f;
  for (int c = t; c < N_NODES; c += 256) {
    const float* kr = k + c * 64;
    float s = 0.f;
    #pragma unroll
    for (int j = 0; j < 64; ++j) s = fmaf(qrow[j], kr[j], s);
    s *= 0.125f;
    sc[c] = s;
    lmax = fmaxf(lmax, s);
  }
  red[t] = lmax; __syncthreads();
  for (int off = 128; off > 0; off >>= 1) { if (t < off) red[t] = fmaxf(red[t], red[t + off]); __syncthreads(); }
  float m = red[0]; __syncthreads();
  float lsum = 0.f;
  for (int c = t; c < N_NODES; c += 256) { float e = __expf(sc[c] - m); sc[c] = e; lsum += e; }
  red[t] = lsum; __syncthreads();
  for (int off = 128; off > 0; off >>= 1) { if (t < off) red[t] += red[t + off]; __syncthreads(); }
  float inv = 1.0f / red[0];
  for (int c = t; c < N_NODES; c += 256) attn[(size_t)n * N_NODES + c] = sc[c] * inv;
}

// ---------------- pack X[node][b*66+i] (zero-padded to LDX cols) ----------------
__global__ void pack_x_kernel(const float* __restrict__ inputs, const float* __restrict__ state,
                              const float* __restrict__ r, int use_r, float* __restrict__ X) {
  int idx = blockIdx.x * blockDim.x + threadIdx.x;     // N_NODES*LDX
  int n = idx / LDX, col = idx % LDX;
  float v = 0.f;
  if (col < NCOLS) {
    int b = col / FEAT, i = col % FEAT;
    int row = b * N_NODES + n;
    if (i < IN_DIM) {
      v = inputs[(size_t)row * IN_DIM + i];
    } else {
      float h = state[(size_t)row * UNITS_ + (i - IN_DIM)];
      v = use_r ? r[(size_t)row * UNITS_ + (i - IN_DIM)] * h : h;
    }
  }
  X[idx] = v;
}

// ---------------- pack gate weights into K-paired layout Wp[k/2][col][2] ----------------
__global__ void pack_w_kernel(const float* __restrict__ rK, const float* __restrict__ uK,
                              const float* __restrict__ cK, float* __restrict__ Wp) {
  int idx = blockIdx.x * blockDim.x + threadIdx.x;   // 3*264*64 = 50688
  if (idx >= 3 * KROWS * UNITS_) return;
  int w = idx / (KROWS * UNITS_), rem = idx % (KROWS * UNITS_);
  int k = rem / UNITS_, c = rem % UNITS_;
  const float* src = (w == 0) ? rK : (w == 1) ? uK : cK;
  Wp[w * KROWS * UNITS_ + (k >> 1) * 128 + c * 2 + (k & 1)] = src[k * UNITS_ + c];
}

// ---------------- support GEMM: Y_z = S_z @ X  (M=1024, K=1024, N=LDX), f32 WMMA ----------------
__global__ __launch_bounds__(256) void support_gemm_kernel(
    const float* __restrict__ S0, const float* __restrict__ S1, const float* __restrict__ S2,
    const float* __restrict__ X,
    float* __restrict__ Y0, float* __restrict__ Y1, float* __restrict__ Y2) {
  __shared__ float As[64 * 18];     // 64 rows x 16 k, stride 18 (keeps b64 align, spreads banks)
  __shared__ float Bs[8 * 256];     // [kpair 0..7][col 0..127][2] : K-pair contiguous per column
  const float* S = (blockIdx.z == 0) ? S0 : (blockIdx.z == 1) ? S1 : S2;
  float* Y = (blockIdx.z == 0) ? Y0 : (blockIdx.z == 1) ? Y1 : Y2;

  const int tid  = threadIdx.x;
  const int wave = tid >> 5, lane = tid & 31;
  const int wm = wave & 3, wn = wave >> 2;     // 4x2 wave grid -> 64x128 block tile
  const int m0 = blockIdx.y * 64;
  const int n0 = blockIdx.x * 128;
  const int lr = lane & 15;
  const int koff = (lane >> 4) * 2;            // A/B frag: lanes 0-15 K0/K1, 16-31 K2/K3

  v8f acc0 = {}, acc1 = {}, acc2 = {}, acc3 = {};

  const int arow = tid >> 2;                   // 0..63
  const int acol = (tid & 3) * 4;              // 0,4,8,12
  const int nb2 = (wn * 64 + lr) * 2;          // paired-col base in Bs

  for (int k0 = 0; k0 < N_NODES; k0 += 16) {
    float4 va = *(const float4*)(S + (size_t)(m0 + arow) * N_NODES + k0 + acol);
    As[arow * 18 + acol + 0] = va.x;
    As[arow * 18 + acol + 1] = va.y;
    As[arow * 18 + acol + 2] = va.z;
    As[arow * 18 + acol + 3] = va.w;
    #pragma unroll
    for (int s = 0; s < 2; ++s) {
      int slot = tid + s * 256;                // 0..511 float4 slots
      int brow = slot >> 5;                    // k within chunk, 0..15
      int bcol = (slot & 31) * 4;              // 0..124
      float4 vb = *(const float4*)(X + (size_t)(k0 + brow) * LDX + n0 + bcol);
      float* bp = &Bs[(brow >> 1) * 256 + bcol * 2 + (brow & 1)];
      bp[0] = vb.x; bp[2] = vb.y; bp[4] = vb.z; bp[6] = vb.w;
    }
    __syncthreads();
    #pragma unroll
    for (int kk = 0; kk < 16; kk += 4) {
      v2f a = *(const v2f*)&As[(wm * 16 + lr) * 18 + kk + koff];
      const float* brow_ = &Bs[((kk + koff) >> 1) * 256 + nb2];
      v2f b0 = *(const v2f*)(brow_ +  0);      // cols +0
      v2f b1 = *(const v2f*)(brow_ + 32);      // cols +16
      v2f b2 = *(const v2f*)(brow_ + 64);      // cols +32
      v2f b3 = *(const v2f*)(brow_ + 96);      // cols +48
      acc0 = __builtin_amdgcn_wmma_f32_16x16x4_f32(false, a, false, b0, (short)0, acc0, false, false);
      acc1 = __builtin_amdgcn_wmma_f32_16x16x4_f32(false, a, false, b1, (short)0, acc1, false, false);
      acc2 = __builtin_amdgcn_wmma_f32_16x16x4_f32(false, a, false, b2, (short)0, acc2, false, false);
      acc3 = __builtin_amdgcn_wmma_f32_16x16x4_f32(false, a, false, b3, (short)0, acc3, false, false);
    }
    __syncthreads();
  }
  // D layout: lanes 0-15 -> M=j, lanes 16-31 -> M=j+8; N = lane&15 within 16-col block
  const int nbase = n0 + wn * 64 + lr;
  const int mbase = m0 + wm * 16 + (lane >> 4) * 8;
  #pragma unroll
  for (int j = 0; j < 8; ++j) {
    float* yr = Y + (size_t)(mbase + j) * LDX + nbase;
    yr[0]  = acc0[j];
    yr[16] = acc1[j];
    yr[32] = acc2[j];
    yr[48] = acc3[j];
  }
}

// ---------------- pack stacked activations Ast[row=b*N+n][i*4+m] ----------------
__global__ void pack_astack_kernel(const float* __restrict__ X, const float* __restrict__ Y0,
                                   const float* __restrict__ Y1, const float* __restrict__ Y2,
                                   float* __restrict__ A) {
  int idx = blockIdx.x * blockDim.x + threadIdx.x;   // NROWS*KROWS
  int row = idx / KROWS, k = idx % KROWS;
  int b = row >> 10, n = row & 1023;
  int i = k >> 2, m = k & 3;
  const float* src = (m == 0) ? X : (m == 1) ? Y0 : (m == 2) ? Y1 : Y2;
  A[idx] = src[(size_t)n * LDX + b * FEAT + i];
}

// ---------------- gate GEMM: [NROWS x 264] @ [264 x 64], fused bias + activation ----------------
// W is in K-paired layout Wp[k/2][col][2].
// mode 0: out = sigmoid(v)   (r and u gates)
// mode 1: out = u*h + (1-u)*tanh(v)   (c gate + GRU combine -> final h)
__global__ __launch_bounds__(256) void gate_gemm_kernel(
    const float* __restrict__ A, const float* __restrict__ Wp, const float* __restrict__ bias,
    const float* __restrict__ ubuf, const float* __restrict__ hprev,
    float* __restrict__ out, int mode) {
  const int tid  = threadIdx.x;
  const int wave = tid >> 5, lane = tid & 31;
  const int wm = wave >> 2, wn = wave & 3;            // 2x4 wave grid -> 32 rows x 64 cols
  const int row0 = blockIdx.x * 32 + wm * 16;
  const int lr = lane & 15;
  const int koff = (lane >> 4) * 2;
  const int col = wn * 16 + lr;

  v8f acc = {};
  #pragma unroll 2
  for (int k = 0; k < KROWS; k += 4) {
    v2f a = *(const v2f*)(A + (size_t)(row0 + lr) * KROWS + k + koff);
    v2f b = *(const v2f*)(Wp + ((k + koff) >> 1) * 128 + col * 2);
    acc = __builtin_amdgcn_wmma_f32_16x16x4_f32(false, a, false, b, (short)0, acc, false, false);
  }
  const float bv = bias[col];
  const int mb = row0 + (lane >> 4) * 8;
  #pragma unroll
  for (int j = 0; j < 8; ++j) {
    int rr = mb + j;
    float v = acc[j] + bv;
    size_t oi = (size_t)rr * UNITS_ + col;
    if (mode == 0) {
      out[oi] = 1.0f / (1.0f + __expf(-v));
    } else {
      float uu = ubuf[oi];
      float h  = hprev[oi];
      out[oi] = uu * h + (1.0f - uu) * tanhf(v);
    }
  }
}

extern "C" void kernel_launch(void* const* d_in, const int* in_sizes, int n_in,
                              void* d_out, int out_size, void* d_ws, size_t ws_size,
                              hipStream_t stream) {
  const float* inputs = (const float*)d_in[0];
  const float* state  = (const float*)d_in[1];
  const float* adj0   = (const float*)d_in[2];
  const float* adj1   = (const float*)d_in[3];
  const float* feat0  = (const float*)d_in[4];
  const float* Wq     = (const float*)d_in[5];
  const float* Wk     = (const float*)d_in[6];
  const float* rK     = (const float*)d_in[7];
  const float* rB     = (const float*)d_in[8];
  const float* uK     = (const float*)d_in[9];
  const float* uB     = (const float*)d_in[10];
  const float* cK     = (const float*)d_in[11];
  const float* cB     = (const float*)d_in[12];
  float* out = (float*)d_out;
  float* ws  = (float*)d_ws;

  float* q    = ws + OFF_Q;
  float* kbuf = ws + OFF_K;
  float* attn = ws + OFF_ATTN;
  float* X    = ws + OFF_X;
  float* Y0   = ws + OFF_Y0;
  float* Y1   = ws + OFF_Y1;
  float* Y2   = ws + OFF_Y2;
  float* Ast  = ws + OFF_AST;
  float* rb   = ws + OFF_R;
  float* ub   = ws + OFF_U;
  float* Wp   = ws + OFF_WP;

  // attention support + weight re-layout
  qk_kernel<<<(N_NODES * UNITS_) / 256, 256, 0, stream>>>(feat0, Wq, Wk, q, kbuf);
  attn_kernel<<<N_NODES, 256, 0, stream>>>(q, kbuf, attn);
  pack_w_kernel<<<(3 * KROWS * UNITS_ + 255) / 256, 256, 0, stream>>>(rK, uK, cK, Wp);

  dim3 sg(LDX / 128, N_NODES / 64, 3);
  const int WSZ = KROWS * UNITS_;

  // phase 1: r and u gates (shared diffusion products)
  pack_x_kernel<<<(N_NODES * LDX) / 256, 256, 0, stream>>>(inputs, state, nullptr, 0, X);
  support_gemm_kernel<<<sg, 256, 0, stream>>>(adj0, adj1, attn, X, Y0, Y1, Y2);
  pack_astack_kernel<<<(NROWS * KROWS) / 256, 256, 0, stream>>>(X, Y0, Y1, Y2, Ast);
  gate_gemm_kernel<<<NROWS / 32, 256, 0, stream>>>(Ast, Wp + 0 * WSZ, rB, nullptr, nullptr, rb, 0);
  gate_gemm_kernel<<<NROWS / 32, 256, 0, stream>>>(Ast, Wp + 1 * WSZ, uB, nullptr, nullptr, ub, 0);

  // phase 2: c gate on r*h, fused GRU combine -> out
  pack_x_kernel<<<(N_NODES * LDX) / 256, 256, 0, stream>>>(inputs, state, rb, 1, X);
  support_gemm_kernel<<<sg, 256, 0, stream>>>(adj0, adj1, attn, X, Y0, Y1, Y2);
  pack_astack_kernel<<<(NROWS * KROWS) / 256, 256, 0, stream>>>(X, Y0, Y1, Y2, Ast);
  gate_gemm_kernel<<<NROWS / 32, 256, 0, stream>>>(Ast, Wp + 2 * WSZ, cB, ub, state, out, 1);
}